// get_model_32882269618443
// MI455X (gfx1250) — compile-verified
//
#include <hip/hip_runtime.h>
#include <hip/hip_bf16.h>
#include <math.h>

// ---------------- model constants ----------------
constexpr int kB   = 16;    // batches
constexpr int kN   = 2048;  // points
constexpr int kG   = 128;   // groups / sequence length
constexpr int kM   = 32;    // neighbors per group
constexpr int kDM  = 384;   // model dim
constexpr int kNL  = 12;    // mixer layers
constexpr int kDI  = 768;   // inner dim
constexpr int kDS  = 16;    // state dim
constexpr int kDTR = 24;    // dt rank
constexpr float kEPS = 1e-5f;

typedef __attribute__((ext_vector_type(2))) float v2f;
typedef __attribute__((ext_vector_type(8))) float v8f;

// Full-precision CDNA5 matrix op: D(16x16,f32) = A(16x4,f32) x B(4x16,f32) + C
__device__ __forceinline__ v8f wmma4(v2f a, v2f b, v8f c) {
  return __builtin_amdgcn_wmma_f32_16x16x4_f32(false, a, false, b, (short)0, c,
                                               false, false);
}

__device__ __forceinline__ float siluf(float x) { return x / (1.f + expf(-x)); }
__device__ __forceinline__ float softplusf(float x) {
  return fmaxf(x, 0.f) + log1pf(expf(-fabsf(x)));
}
__device__ __forceinline__ float geluf(float x) {
  float x3 = x * x * x;
  return 0.5f * x * (1.f + tanhf(0.7978845608028654f * (x + 0.044715f * x3)));
}

// =====================================================================
// 1) Farthest point sampling: one block per batch, sequential over G.
// =====================================================================
__global__ void fps_kernel(const float* __restrict__ xyz, int* __restrict__ cidx) {
  int b = blockIdx.x;
  const float* pts = xyz + (size_t)b * kN * 3;
  int t = threadIdx.x;  // 256 threads, 8 points each
  float px[8], py[8], pz[8], dmin[8];
#pragma unroll
  for (int i = 0; i < 8; ++i) {
    int p = t + i * 256;
    px[i] = pts[p * 3 + 0];
    py[i] = pts[p * 3 + 1];
    pz[i] = pts[p * 3 + 2];
    dmin[i] = 1e10f;
  }
  __shared__ float rv[256];
  __shared__ int ri[256];
  __shared__ float cx, cy, cz;
  __shared__ int cur;
  if (t == 0) cur = 0;
  __syncthreads();
  for (int g = 0; g < kG; ++g) {
    if (t == 0) {
      cidx[b * kG + g] = cur;
      cx = pts[cur * 3 + 0];
      cy = pts[cur * 3 + 1];
      cz = pts[cur * 3 + 2];
    }
    __syncthreads();
    float bestv = -1.f;
    int besti = kN;
#pragma unroll
    for (int i = 0; i < 8; ++i) {
      float dx = px[i] - cx, dy = py[i] - cy, dz = pz[i] - cz;
      float dd = dx * dx + dy * dy + dz * dz;
      dmin[i] = fminf(dmin[i], dd);
      int p = t + i * 256;
      if (dmin[i] > bestv || (dmin[i] == bestv && p < besti)) {
        bestv = dmin[i];
        besti = p;
      }
    }
    rv[t] = bestv;
    ri[t] = besti;
    __syncthreads();
    for (int s = 128; s > 0; s >>= 1) {
      if (t < s) {
        float v2 = rv[t + s];
        int i2 = ri[t + s];
        if (v2 > rv[t] || (v2 == rv[t] && i2 < ri[t])) {
          rv[t] = v2;
          ri[t] = i2;
        }
      }
      __syncthreads();
    }
    if (t == 0) cur = ri[0];
    __syncthreads();
  }
}

// =====================================================================
// 2) kNN grouping: one block per (b,g). Order within the 32 selected
//    neighbors is irrelevant (results are max-pooled downstream).
// =====================================================================
__global__ void group_kernel(const float* __restrict__ xyz,
                             const int* __restrict__ cidx,
                             float* __restrict__ nb, float* __restrict__ centerOut) {
  int gid = blockIdx.x;
  int b = gid / kG;
  const float* pts = xyz + (size_t)b * kN * 3;
  int ci = cidx[gid];
  __shared__ float d2s[kN];
  __shared__ float rv[256];
  __shared__ int ri[256];
  float cx = pts[ci * 3 + 0], cy = pts[ci * 3 + 1], cz = pts[ci * 3 + 2];
  int t = threadIdx.x;
#pragma unroll
  for (int i = 0; i < 8; ++i) {
    int p = t + i * 256;
    float dx = pts[p * 3 + 0] - cx, dy = pts[p * 3 + 1] - cy, dz = pts[p * 3 + 2] - cz;
    d2s[p] = dx * dx + dy * dy + dz * dz;
  }
  if (t == 0) {
    centerOut[gid * 3 + 0] = cx;
    centerOut[gid * 3 + 1] = cy;
    centerOut[gid * 3 + 2] = cz;
  }
  __syncthreads();
  for (int m = 0; m < kM; ++m) {
    float bv = 1e30f;
    int bi = kN;
#pragma unroll
    for (int i = 0; i < 8; ++i) {
      int p = t + i * 256;
      float v = d2s[p];
      if (v < bv || (v == bv && p < bi)) {
        bv = v;
        bi = p;
      }
    }
    rv[t] = bv;
    ri[t] = bi;
    __syncthreads();
    for (int s = 128; s > 0; s >>= 1) {
      if (t < s) {
        float v2 = rv[t + s];
        int i2 = ri[t + s];
        if (v2 < rv[t] || (v2 == rv[t] && i2 < ri[t])) {
          rv[t] = v2;
          ri[t] = i2;
        }
      }
      __syncthreads();
    }
    if (t == 0) {
      int p = ri[0];
      nb[(size_t)(gid * kM + m) * 3 + 0] = pts[p * 3 + 0] - cx;
      nb[(size_t)(gid * kM + m) * 3 + 1] = pts[p * 3 + 1] - cy;
      nb[(size_t)(gid * kM + m) * 3 + 2] = pts[p * 3 + 2] - cz;
      d2s[p] = 1e30f;
    }
    __syncthreads();
  }
}

// =====================================================================
// 3) Per-group encoder MLP. One 256-thread block (8 waves) per group.
//    WMMA f32 16x16x4 fragments:
//      A: lane (m = l&15, kh = l>>4), a[j] = A[m][k0 + 2*kh + j]
//      B: lane (n = l&15, kh),        b[j] = W[n][k0 + 2*kh + j]   (W is (N,K))
//      C: vgpr r, lane l -> row = r + 8*(l>>4), col = l&15
// =====================================================================
__device__ inline void lds_gemm(const float* __restrict__ As, int lda,
                                const float* __restrict__ W,
                                const float* __restrict__ bias,
                                float* __restrict__ Cs, int ldc, int Ncols, int K) {
  int wave = threadIdx.x >> 5, lane = threadIdx.x & 31;
  int m = lane & 15, kh = lane >> 4;
  int ntile = 2 * (Ncols >> 4);  // 2 row-tiles (M=32)
  for (int tile = wave; tile < ntile; tile += 8) {
    int mt = tile & 1, nt = tile >> 1;
    const float* Arow = As + (size_t)(mt * 16 + m) * lda + 2 * kh;
    int n = nt * 16 + m;
    const float* Wrow = W + (size_t)n * K + 2 * kh;
    v8f acc = {};
    for (int k0 = 0; k0 < K; k0 += 4) {
      v2f a, bv;
      a[0] = Arow[k0];
      a[1] = Arow[k0 + 1];
      bv[0] = Wrow[k0];
      bv[1] = Wrow[k0 + 1];
      acc = wmma4(a, bv, acc);
    }
    float bb = bias[n];
#pragma unroll
    for (int r = 0; r < 8; ++r) {
      Cs[(size_t)(mt * 16 + r + 8 * kh) * ldc + n] = acc[r] + bb;
    }
  }
}

// A = [fg broadcast (256) | f2 (32x256)]   -> relu(bn2(A @ e2w1^T + e2b1))
__device__ inline void gemm_concat(const float* __restrict__ fg,
                                   const float* __restrict__ f2,
                                   const float* __restrict__ W,
                                   const float* __restrict__ bias,
                                   const float* __restrict__ bng,
                                   const float* __restrict__ bnb,
                                   float* __restrict__ Cs) {
  int wave = threadIdx.x >> 5, lane = threadIdx.x & 31;
  int m = lane & 15, kh = lane >> 4;
  const float BNS = rsqrtf(1.f + kEPS);
  for (int tile = wave; tile < 64; tile += 8) {  // 2 x 32 tiles, N=512, K=512
    int mt = tile & 1, nt = tile >> 1;
    int row = mt * 16 + m;
    int n = nt * 16 + m;
    const float* Wrow = W + (size_t)n * 512 + 2 * kh;
    v8f acc = {};
    for (int k0 = 0; k0 < 512; k0 += 4) {
      int k = k0 + 2 * kh;  // always even
      v2f a, bv;
      a[0] = (k < 256) ? fg[k] : f2[(size_t)row * 256 + (k - 256)];
      a[1] = (k + 1 < 256) ? fg[k + 1] : f2[(size_t)row * 256 + (k + 1 - 256)];
      bv[0] = Wrow[k0];
      bv[1] = Wrow[k0 + 1];
      acc = wmma4(a, bv, acc);
    }
    float bb = bias[n];
    float g = bng[n] * BNS;
    float b2 = bnb[n];
#pragma unroll
    for (int r = 0; r < 8; ++r) {
      float v = (acc[r] + bb) * g + b2;
      Cs[(size_t)(mt * 16 + r + 8 * kh) * 512 + n] = fmaxf(v, 0.f);
    }
  }
}

__global__ void encoder_kernel(const float* __restrict__ nb,
                               const float* __restrict__ e1w1, const float* __restrict__ e1b1,
                               const float* __restrict__ bn1g, const float* __restrict__ bn1b,
                               const float* __restrict__ e1w2, const float* __restrict__ e1b2,
                               const float* __restrict__ e2w1, const float* __restrict__ e2b1,
                               const float* __restrict__ bn2g, const float* __restrict__ bn2b,
                               const float* __restrict__ e2w2, const float* __restrict__ e2b2,
                               float* __restrict__ hout) {
  extern __shared__ float sm[];
  float* sx = sm;                 // 96 (padded to 128)
  float* f1 = sm + 128;           // 32x128
  float* f2 = f1 + 32 * 128;      // 32x256
  float* fg = f2 + 32 * 256;      // 256
  float* f3 = fg + 256;           // 32x512
  float* f5 = f1;                 // 32x384 reuses f1+f2 region (12288 floats)

  int gid = blockIdx.x, t = threadIdx.x;
  for (int i = t; i < 96; i += 256) sx[i] = nb[(size_t)gid * 96 + i];
  __syncthreads();

  const float BNS = rsqrtf(1.f + kEPS);
  // f1 = relu(bn1(x @ e1w1^T + e1b1)) : 32x128, K=3 (scalar, trivial FLOPs)
  for (int idx = t; idx < 32 * 128; idx += 256) {
    int m = idx >> 7, n = idx & 127;
    float s = sx[m * 3 + 0] * e1w1[n * 3 + 0] + sx[m * 3 + 1] * e1w1[n * 3 + 1] +
              sx[m * 3 + 2] * e1w1[n * 3 + 2] + e1b1[n];
    s = s * (bn1g[n] * BNS) + bn1b[n];
    f1[idx] = fmaxf(s, 0.f);
  }
  __syncthreads();
  // f2 = f1 @ e1w2^T + e1b2 : 32x256, K=128  (WMMA)
  lds_gemm(f1, 128, e1w2, e1b2, f2, 256, 256, 128);
  __syncthreads();
  // fg = column max of f2
  for (int n = t; n < 256; n += 256) {
    float mx = f2[n];
#pragma unroll
    for (int m = 1; m < 32; ++m) mx = fmaxf(mx, f2[(size_t)m * 256 + n]);
    fg[n] = mx;
  }
  __syncthreads();
  // f3 = relu(bn2([fg|f2] @ e2w1^T + e2b1)) : 32x512, K=512  (WMMA)
  gemm_concat(fg, f2, e2w1, e2b1, bn2g, bn2b, f3);
  __syncthreads();
  // f5 = f3 @ e2w2^T + e2b2 : 32x384, K=512  (WMMA)
  lds_gemm(f3, 512, e2w2, e2b2, f5, 384, 384, 512);
  __syncthreads();
  // tokens = column max of f5
  for (int n = t; n < 384; n += 256) {
    float mx = f5[n];
#pragma unroll
    for (int m = 1; m < 32; ++m) mx = fmaxf(mx, f5[(size_t)m * 384 + n]);
    hout[(size_t)gid * kDM + n] = mx;
  }
}

// =====================================================================
// 4) pos MLP: h += gelu(center @ pw1^T + pb1) @ pw2^T + pb2
// =====================================================================
__global__ void pos_kernel(const float* __restrict__ center,
                           const float* __restrict__ pw1, const float* __restrict__ pb1,
                           const float* __restrict__ pw2, const float* __restrict__ pb2,
                           float* __restrict__ hbuf) {
  __shared__ float hid[128];
  int row = blockIdx.x, t = threadIdx.x;  // 128 threads
  float c0 = center[row * 3 + 0], c1 = center[row * 3 + 1], c2 = center[row * 3 + 2];
  float u = c0 * pw1[t * 3 + 0] + c1 * pw1[t * 3 + 1] + c2 * pw1[t * 3 + 2] + pb1[t];
  hid[t] = geluf(u);
  __syncthreads();
  for (int j = t; j < kDM; j += 128) {
    float s = pb2[j];
    for (int k = 0; k < 128; ++k) s += hid[k] * pw2[(size_t)j * 128 + k];
    hbuf[(size_t)row * kDM + j] += s;
  }
}

// =====================================================================
// 5) Generic row-major GEMM: C(RxN) = act(A(RxK) @ W(N,K)^T + bias)
//    One 16x16 tile per wave, f32 WMMA (16x16x4), K multiple of 4.
//    act: 0 = none, 1 = softplus
// =====================================================================
__global__ void gemm_kernel(const float* __restrict__ A, int lda,
                            const float* __restrict__ W,
                            const float* __restrict__ bias,
                            float* __restrict__ C, int ldc,
                            int R, int Ncols, int K, int act) {
  int wave = threadIdx.x >> 5, lane = threadIdx.x & 31;
  int m = lane & 15, kh = lane >> 4;
  int ntn = (Ncols + 15) >> 4;
  int tiles = (R >> 4) * ntn;
  int tile = blockIdx.x * 8 + wave;
  if (tile >= tiles) return;
  int mt = tile / ntn, nt = tile % ntn;
  const float* Arow = A + (size_t)(mt * 16 + m) * lda + 2 * kh;
  int n = nt * 16 + m;
  bool nv = n < Ncols;
  const float* Wrow = W + (size_t)(nv ? n : 0) * K + 2 * kh;
  v8f acc = {};
  for (int k0 = 0; k0 < K; k0 += 4) {
    v2f a, bv;
    a[0] = Arow[k0];
    a[1] = Arow[k0 + 1];
    float w0 = Wrow[k0], w1 = Wrow[k0 + 1];
    bv[0] = nv ? w0 : 0.f;
    bv[1] = nv ? w1 : 0.f;
    acc = wmma4(a, bv, acc);
  }
  float bb = (bias != nullptr && nv) ? bias[n] : 0.f;
#pragma unroll
  for (int r = 0; r < 8; ++r) {
    float v = acc[r] + bb;
    if (act == 1) v = softplusf(v);
    if (nv) C[(size_t)(mt * 16 + r + 8 * kh) * ldc + n] = v;
  }
}

// =====================================================================
// 6) Causal depthwise conv (k=4) + silu. One thread per (b,d).
// =====================================================================
__global__ void conv_kernel(const float* __restrict__ xraw, int ldx,
                            const float* __restrict__ cw, const float* __restrict__ cb,
                            float* __restrict__ xout) {
  int tid = blockIdx.x * blockDim.x + threadIdx.x;
  if (tid >= kB * kDI) return;
  int b = tid / kDI, d = tid % kDI;
  float w0 = cw[d * 4 + 0], w1 = cw[d * 4 + 1], w2 = cw[d * 4 + 2], w3 = cw[d * 4 + 3];
  float bias = cb[d];
  float x0 = 0.f, x1 = 0.f, x2 = 0.f;
  const float* src = xraw + (size_t)b * kG * ldx + d;
  float* dst = xout + (size_t)b * kG * kDI + d;
  for (int l = 0; l < kG; ++l) {
    float x3 = src[(size_t)l * ldx];
    float s = x0 * w0 + x1 * w1 + x2 * w2 + x3 * w3 + bias;
    dst[(size_t)l * kDI] = siluf(s);
    x0 = x1;
    x1 = x2;
    x2 = x3;
  }
}

// =====================================================================
// 7) Selective scan. One thread per (b,d); state h[16] in registers,
//    B/C rows for the current (b,l) broadcast through LDS.
//    ypre = (scan_y + Dp*xin) * silu(z)
// =====================================================================
__global__ void scan_kernel(const float* __restrict__ dt, const float* __restrict__ xin,
                            const float* __restrict__ dbl, const float* __restrict__ xz,
                            const float* __restrict__ A_log, const float* __restrict__ Dp,
                            float* __restrict__ ypre) {
  __shared__ float sB[kDS], sC[kDS];
  int blk = blockIdx.x;  // 48 blocks: 16 batches x 3 d-chunks of 256
  int b = blk / 3;
  int d = (blk % 3) * 256 + threadIdx.x;
  float Ad[kDS], h[kDS];
#pragma unroll
  for (int s = 0; s < kDS; ++s) {
    Ad[s] = -expf(A_log[(size_t)d * kDS + s]);
    h[s] = 0.f;
  }
  float Dv = Dp[d];
  for (int l = 0; l < kG; ++l) {
    size_t row = (size_t)b * kG + l;
    if (threadIdx.x < 2 * kDS) {
      float v = dbl[row * 56 + kDTR + threadIdx.x];
      if (threadIdx.x < kDS)
        sB[threadIdx.x] = v;
      else
        sC[threadIdx.x - kDS] = v;
    }
    __syncthreads();
    float dtv = dt[row * kDI + d];
    float xv = xin[row * kDI + d];
    float dx = dtv * xv;
    float y = 0.f;
#pragma unroll
    for (int s = 0; s < kDS; ++s) {
      float dA = expf(dtv * Ad[s]);
      h[s] = dA * h[s] + dx * sB[s];
      y += h[s] * sC[s];
    }
    y += Dv * xv;
    float zv = xz[row * (2 * kDI) + kDI + d];
    ypre[row * kDI + d] = y * siluf(zv);
    __syncthreads();
  }
}

// =====================================================================
// 8) Residual update + LayerNorm.
//    mode 0: res = h;  mode 1: res += h;  mode 2: final (x = h+res, no store)
// =====================================================================
__global__ void ln_kernel(const float* __restrict__ hin, float* __restrict__ residual,
                          const float* __restrict__ w, const float* __restrict__ bvec,
                          float* __restrict__ outp, int mode) {
  int row = blockIdx.x, t = threadIdx.x;  // 128 threads x 3 elems
  __shared__ float red[128];
  float v[3];
#pragma unroll
  for (int i = 0; i < 3; ++i) {
    int j = t * 3 + i;
    float x = hin[(size_t)row * kDM + j];
    if (mode != 0) x += residual[(size_t)row * kDM + j];
    if (mode != 2) residual[(size_t)row * kDM + j] = x;
    v[i] = x;
  }
  red[t] = v[0] + v[1] + v[2];
  __syncthreads();
  for (int s = 64; s > 0; s >>= 1) {
    if (t < s) red[t] += red[t + s];
    __syncthreads();
  }
  float mu = red[0] / (float)kDM;
  __syncthreads();
  float sv = 0.f;
#pragma unroll
  for (int i = 0; i < 3; ++i) {
    float dd = v[i] - mu;
    sv += dd * dd;
  }
  red[t] = sv;
  __syncthreads();
  for (int s = 64; s > 0; s >>= 1) {
    if (t < s) red[t] += red[t + s];
    __syncthreads();
  }
  float inv = rsqrtf(red[0] / (float)kDM + kEPS);
#pragma unroll
  for (int i = 0; i < 3; ++i) {
    int j = t * 3 + i;
    outp[(size_t)row * kDM + j] = (v[i] - mu) * inv * w[j] + bvec[j];
  }
}

// =====================================================================
// Host driver
// =====================================================================
extern "C" void kernel_launch(void* const* d_in, const int* in_sizes, int n_in,
                              void* d_out, int out_size, void* d_ws, size_t ws_size,
                              hipStream_t stream) {
  (void)in_sizes; (void)n_in; (void)out_size; (void)ws_size;
  const float* xyz    = (const float*)d_in[0];
  const float* e1w1   = (const float*)d_in[1];
  const float* e1b1   = (const float*)d_in[2];
  const float* bn1g   = (const float*)d_in[3];
  const float* bn1b   = (const float*)d_in[4];
  const float* e1w2   = (const float*)d_in[5];
  const float* e1b2   = (const float*)d_in[6];
  const float* e2w1   = (const float*)d_in[7];
  const float* e2b1   = (const float*)d_in[8];
  const float* bn2g   = (const float*)d_in[9];
  const float* bn2b   = (const float*)d_in[10];
  const float* e2w2   = (const float*)d_in[11];
  const float* e2b2   = (const float*)d_in[12];
  const float* pw1    = (const float*)d_in[13];
  const float* pb1    = (const float*)d_in[14];
  const float* pw2    = (const float*)d_in[15];
  const float* pb2    = (const float*)d_in[16];
  const float* in_w   = (const float*)d_in[17];
  const float* conv_w = (const float*)d_in[18];
  const float* conv_b = (const float*)d_in[19];
  const float* xp_w   = (const float*)d_in[20];
  const float* dt_w   = (const float*)d_in[21];
  const float* dt_b   = (const float*)d_in[22];
  const float* A_log  = (const float*)d_in[23];
  const float* Dp     = (const float*)d_in[24];
  const float* out_w  = (const float*)d_in[25];
  const float* lnw    = (const float*)d_in[26];
  const float* lnb    = (const float*)d_in[27];
  const float* fnw    = (const float*)d_in[28];
  const float* fnb    = (const float*)d_in[29];

  const int ROWS = kB * kG;  // 2048

  // ---- workspace carve-up (floats) ----
  float* wsf  = (float*)d_ws;
  int* cidx   = (int*)d_ws;
  size_t off  = kB * kG;                                  // cidx (int)
  float* center = wsf + off; off += (size_t)ROWS * 3;
  float* nb     = wsf + off; off += (size_t)ROWS * kM * 3;
  float* hbuf   = wsf + off; off += (size_t)ROWS * kDM;
  float* resid  = wsf + off; off += (size_t)ROWS * kDM;
  float* xln    = wsf + off; off += (size_t)ROWS * kDM;
  float* xz     = wsf + off; off += (size_t)ROWS * 2 * kDI;
  float* xin    = wsf + off; off += (size_t)ROWS * kDI;
  float* dblb   = wsf + off; off += (size_t)ROWS * (kDTR + 2 * kDS);
  float* dtb    = wsf + off; off += (size_t)ROWS * kDI;
  float* ypre   = wsf + off; off += (size_t)ROWS * kDI;

  // ---- grouping ----
  fps_kernel<<<kB, 256, 0, stream>>>(xyz, cidx);
  group_kernel<<<ROWS, 256, 0, stream>>>(xyz, cidx, nb, center);

  // ---- encoder (dynamic LDS: 29056 floats ~ 114 KB, well under 320 KB/WGP) ----
  const size_t ENC_SM = (size_t)(128 + 32 * 128 + 32 * 256 + 256 + 32 * 512) * sizeof(float);
  hipFuncSetAttribute(reinterpret_cast<const void*>(encoder_kernel),
                      hipFuncAttributeMaxDynamicSharedMemorySize, (int)ENC_SM);
  encoder_kernel<<<ROWS, 256, ENC_SM, stream>>>(nb, e1w1, e1b1, bn1g, bn1b, e1w2, e1b2,
                                                e2w1, e2b1, bn2g, bn2b, e2w2, e2b2, hbuf);
  pos_kernel<<<ROWS, 128, 0, stream>>>(center, pw1, pb1, pw2, pb2, hbuf);

  auto gemm_blocks = [](int R, int Ncols) {
    int tiles = (R / 16) * ((Ncols + 15) / 16);
    return (tiles + 7) / 8;
  };

  // ---- mixer layers ----
  for (int l = 0; l < kNL; ++l) {
    const float* in_w_l   = in_w + (size_t)l * 2 * kDI * kDM;
    const float* conv_w_l = conv_w + (size_t)l * kDI * 4;
    const float* conv_b_l = conv_b + (size_t)l * kDI;
    const float* xp_w_l   = xp_w + (size_t)l * (kDTR + 2 * kDS) * kDI;
    const float* dt_w_l   = dt_w + (size_t)l * kDI * kDTR;
    const float* dt_b_l   = dt_b + (size_t)l * kDI;
    const float* A_log_l  = A_log + (size_t)l * kDI * kDS;
    const float* Dp_l     = Dp + (size_t)l * kDI;
    const float* out_w_l  = out_w + (size_t)l * kDM * kDI;
    const float* lnw_l    = lnw + (size_t)l * kDM;
    const float* lnb_l    = lnb + (size_t)l * kDM;

    // residual update + LN
    ln_kernel<<<ROWS, 128, 0, stream>>>(hbuf, resid, lnw_l, lnb_l, xln, (l == 0) ? 0 : 1);
    // in_proj: (2048x384)x(384x1536)
    gemm_kernel<<<gemm_blocks(ROWS, 2 * kDI), 256, 0, stream>>>(
        xln, kDM, in_w_l, nullptr, xz, 2 * kDI, ROWS, 2 * kDI, kDM, 0);
    // causal conv + silu on xin half of xz
    conv_kernel<<<(kB * kDI + 255) / 256, 256, 0, stream>>>(xz, 2 * kDI, conv_w_l, conv_b_l, xin);
    // x_proj: (2048x768)x(768x56)
    gemm_kernel<<<gemm_blocks(ROWS, kDTR + 2 * kDS), 256, 0, stream>>>(
        xin, kDI, xp_w_l, nullptr, dblb, kDTR + 2 * kDS, ROWS, kDTR + 2 * kDS, kDI, 0);
    // dt = softplus(dbl[:, :24] @ dt_w^T + dt_b): (2048x24)x(24x768)
    gemm_kernel<<<gemm_blocks(ROWS, kDI), 256, 0, stream>>>(
        dblb, kDTR + 2 * kDS, dt_w_l, dt_b_l, dtb, kDI, ROWS, kDI, kDTR, 1);
    // selective scan + gating
    scan_kernel<<<kB * 3, 256, 0, stream>>>(dtb, xin, dblb, xz, A_log_l, Dp_l, ypre);
    // out_proj: (2048x768)x(768x384)
    gemm_kernel<<<gemm_blocks(ROWS, kDM), 256, 0, stream>>>(
        ypre, kDI, out_w_l, nullptr, hbuf, kDM, ROWS, kDM, kDI, 0);
  }

  // ---- final LN(h + residual) -> output ----
  ln_kernel<<<ROWS, 128, 0, stream>>>(hbuf, resid, fnw, fnb, (float*)d_out, 2);
}